// LieCNP_60430189855069
// MI455X (gfx1250) — compile-verified
//
#include <hip/hip_runtime.h>
#include <math.h>

#define BATCH 64
#define NCTX  400
#define NTGT  784
#define GRIDN 28
#define MPTS  784
#define KNN   5
#define HIDW  16
#define CMCO  16

typedef __attribute__((ext_vector_type(2))) float v2f;
typedef __attribute__((ext_vector_type(8))) float v8f;

__device__ __forceinline__ float grid_coord(int i) {
    return -14.0f + (28.0f / 27.0f) * (float)i;
}
#if __has_builtin(__builtin_amdgcn_rcpf)
__device__ __forceinline__ float fast_rcp(float x) { return __builtin_amdgcn_rcpf(x); }
#else
__device__ __forceinline__ float fast_rcp(float x) { return 1.0f / x; }
#endif
__device__ __forceinline__ float swish_f(float x) {
    return x * fast_rcp(1.0f + __expf(-x));    // x * sigmoid(x), v_rcp_f32 path
}
__device__ __forceinline__ float softplus_f(float x) {
    return fmaxf(x, 0.0f) + __logf(1.0f + __expf(-fabsf(x)));
}

// ---- CDNA5 async global->LDS staging (guarded: falls back to plain copies) ----
#if __has_builtin(__builtin_amdgcn_global_load_async_to_lds_b32) && \
    __has_builtin(__builtin_amdgcn_s_wait_asynccnt)
#define ASYNC_LDS 1
typedef __attribute__((address_space(1))) int* gint_p;
typedef __attribute__((address_space(3))) int* lint_p;
__device__ __forceinline__ void cp_b32(const float* g, float* l) {
    __builtin_amdgcn_global_load_async_to_lds_b32((gint_p)g, (lint_p)l, 0, 0);
}
__device__ __forceinline__ void cp_wait() { __builtin_amdgcn_s_wait_asynccnt(0); }
#else
#define ASYNC_LDS 0
__device__ __forceinline__ void cp_b32(const float* g, float* l) { *l = *g; }
__device__ __forceinline__ void cp_wait() {}
#endif

// ---------------------------------------------------------------- KNN (batch-independent)
__global__ void knn_kernel(int* __restrict__ idx) {
    int n = blockIdx.x * blockDim.x + threadIdx.x;
    if (n >= MPTS) return;
    float gx = grid_coord(n / GRIDN), gy = grid_coord(n % GRIDN);
    float bd[KNN]; int bi[KNN];
#pragma unroll
    for (int j = 0; j < KNN; j++) { bd[j] = 3.4e38f; bi[j] = 0; }
    for (int p = 0; p < MPTS; p++) {
        float dx = grid_coord(p / GRIDN) - gx;
        float dy = grid_coord(p % GRIDN) - gy;
        float d2 = dx * dx + dy * dy;
        if (d2 < bd[KNN - 1]) {              // strict '<' keeps lowest-index tie order
            int j = KNN - 1;
            while (j > 0 && d2 < bd[j - 1]) { bd[j] = bd[j - 1]; bi[j] = bi[j - 1]; j--; }
            bd[j] = d2; bi[j] = p;
        }
    }
#pragma unroll
    for (int j = 0; j < KNN; j++) idx[n * KNN + j] = bi[j];
}

// ---------------------------------------------------------------- RBF encode: vals0 = [x, y, h0, h1/(h0+eps)]
__global__ void encode_kernel(const float* __restrict__ cc, const float* __restrict__ cv,
                              const float* __restrict__ psi_ls, const float* __restrict__ psi_os,
                              float* __restrict__ vals) {
    int t = blockIdx.x * blockDim.x + threadIdx.x;
    if (t >= BATCH * MPTS) return;
    int b = t / MPTS, n = t % MPTS;
    float gx = grid_coord(n / GRIDN), gy = grid_coord(n % GRIDN);
    float ls = psi_ls[0], osc = psi_os[0];
    float inv = -0.5f / (ls * ls);
    const float* ccb = cc + (size_t)b * NCTX * 2;
    const float* cvb = cv + (size_t)b * NCTX;
    float h0 = 0.0f, h1 = 0.0f;
    for (int i = 0; i < NCTX; i++) {
        float dx = gx - ccb[2 * i], dy = gy - ccb[2 * i + 1];
        float w = osc * __expf(inv * (dx * dx + dy * dy));
        h0 += w; h1 += w * cvb[i];
    }
    float* o = vals + (size_t)t * 4;
    o[0] = gx; o[1] = gy; o[2] = h0; o[3] = h1 / (h0 + 1e-8f);   // keep precise div here
}

// ---------------------------------------------------------------- per-layer weights (batch-independent)
struct ConvW {
    const float* w1[4]; const float* b1[4];
    const float* w2[4]; const float* b2[4];
    const float* lw[4]; const float* lb[4];
};

__global__ void kern_kernel(ConvW cw, const int* __restrict__ idx, float* __restrict__ kern) {
    int t = blockIdx.x * blockDim.x + threadIdx.x;
    if (t >= 4 * MPTS * KNN) return;
    int li = t / (MPTS * KNN);
    int nk = t % (MPTS * KNN);
    int n = nk / KNN, k = nk % KNN;
    int j = idx[n * KNN + k];
    float dx = grid_coord(j / GRIDN) - grid_coord(n / GRIDN);
    float dy = grid_coord(j % GRIDN) - grid_coord(n % GRIDN);
    const float* w1 = cw.w1[li]; const float* b1 = cw.b1[li];
    const float* w2 = cw.w2[li]; const float* b2 = cw.b2[li];
    float h[HIDW];
#pragma unroll
    for (int q = 0; q < HIDW; q++)
        h[q] = swish_f(dx * w1[q] + dy * w1[HIDW + q] + b1[q]);
    float* out = kern + (size_t)t * CMCO;
#pragma unroll
    for (int m = 0; m < CMCO; m++) {
        float s = b2[m];
#pragma unroll
        for (int q = 0; q < HIDW; q++) s += h[q] * w2[q * CMCO + m];
        out[m] = s;
    }
}

// W[n, kk=(k*Cin+c), o] = sum_m kern[n,k,m] * lw[(c*16+m), o]
__global__ void wfold_kernel(const float* __restrict__ kern, const float* __restrict__ lw,
                             float* __restrict__ W, int Cin, int Cout) {
    int K5 = KNN * Cin;
    int tot = MPTS * K5 * Cout;
    int t = blockIdx.x * blockDim.x + threadIdx.x;
    if (t >= tot) return;
    int o  = t % Cout;
    int kk = (t / Cout) % K5;
    int n  = t / (Cout * K5);
    int k = kk / Cin, c = kk % Cin;
    const float* kr = kern + ((size_t)(n * KNN + k)) * CMCO;
    float s = 0.0f;
#pragma unroll
    for (int m = 0; m < CMCO; m++) s += kr[m] * lw[(c * CMCO + m) * Cout + o];
    W[t] = s;
}

// ---------------------------------------------------------------- WMMA layer: per grid-point GEMM 64 x K5 x Cout
// Templated so K5/Cout are compile-time: K loop fully unrolls into back-to-back
// v_wmma_f32_16x16x4_f32 with immediate-offset ds loads, no branches inside.
template <int Cin, int Cout, bool LAST>
__global__ __launch_bounds__(128)
void layer_kernel(const float* __restrict__ vin, const float* __restrict__ W,
                  const int* __restrict__ idx, const float* __restrict__ lb,
                  float* __restrict__ vout) {
    constexpr int K5 = KNN * Cin;            // 20 / 80 / 160 / 80 (all %4 == 0)
    constexpr int AS = K5 + 4;               // padded A stride: %4==0, bank-spread
    constexpr int NT = (Cout + 15) / 16;     // N tiles
    constexpr int CP = (Cout > 16) ? 40 : 16;// padded W stride: 2*CP mod 64 in [16,48]

    __shared__ float As[BATCH * AS];
    __shared__ float Ws[K5 * CP];

    const int n   = blockIdx.x;
    const int tid = threadIdx.x;

    // zero only padding columns of Ws (never overlaps async-written cells)
    if (16 * NT != Cout) {
        for (int e = tid; e < K5 * CP; e += 128) {
            int o = e % CP;
            if (o >= Cout && o < 16 * NT) Ws[e] = 0.0f;
        }
    }
    // stage A = gathered neighbor values, row b, col kk=(k*Cin+c)
    for (int e = tid; e < BATCH * K5; e += 128) {
        int b = e / K5, kk = e % K5;
        int k = kk / Cin, c = kk % Cin;
        int j = idx[n * KNN + k];
        cp_b32(&vin[((size_t)b * MPTS + j) * Cin + c], &As[b * AS + kk]);
    }
    // stage folded weights for this n (row kk, col o, padded stride CP)
    for (int e = tid; e < K5 * Cout; e += 128) {
        int kk = e / Cout, o = e % Cout;
        cp_b32(&W[(size_t)n * K5 * Cout + e], &Ws[kk * CP + o]);
    }
    cp_wait();
    __syncthreads();

    const int wv   = tid >> 5;        // wave = M-tile (rows 16*wv .. 16*wv+15)
    const int lane = tid & 31;
    const int half = lane >> 4;       // 0: lanes 0-15 (K+0/1), 1: lanes 16-31 (K+2/3)
    const int l16  = lane & 15;
    const int arow = 16 * wv + l16;

#pragma unroll
    for (int ntile = 0; ntile < NT; ntile++) {
        const int col = ntile * 16 + l16;     // < 16*NT <= CP, always safe in LDS
        v8f acc = {0.f, 0.f, 0.f, 0.f, 0.f, 0.f, 0.f, 0.f};
#pragma unroll
        for (int kk0 = 0; kk0 < K5; kk0 += 4) {
            v2f a, bf;
            a.x  = As[arow * AS + kk0 + 2 * half + 0];
            a.y  = As[arow * AS + kk0 + 2 * half + 1];
            bf.x = Ws[(kk0 + 2 * half + 0) * CP + col];
            bf.y = Ws[(kk0 + 2 * half + 1) * CP + col];
            acc = __builtin_amdgcn_wmma_f32_16x16x4_f32(
                false, a, false, bf, (short)0, acc, false, false);
        }
        if (col < Cout) {
            float bias = lb[col];
#pragma unroll
            for (int r = 0; r < 8; r++) {
                int brow = 16 * wv + r + 8 * half;    // batch index
                float v = (acc[r] + bias) * 0.2f;     // (+bias)/K
                if (!LAST)          v = swish_f(v);
                else if (col == 1)  v = softplus_f(v);   // pre-apply softplus to f_sigma
                vout[((size_t)brow * MPTS + n) * Cout + col] = v;
            }
        }
    }
}

// ---------------------------------------------------------------- decode: mu + diagonal Sigma
__global__ void decode_kernel(const float* __restrict__ tc, const float* __restrict__ f,
                              const float* __restrict__ rho_ls, const float* __restrict__ rho_os,
                              float* __restrict__ out_mu, float* __restrict__ out_sigma) {
    int t = blockIdx.x * blockDim.x + threadIdx.x;
    if (t >= BATCH * NTGT) return;
    int b = t / NTGT, q = t % NTGT;
    float tx = tc[(size_t)t * 2 + 0], ty = tc[(size_t)t * 2 + 1];
    float ls = rho_ls[0], osc = rho_os[0];
    float inv = -0.5f / (ls * ls);
    const float* fb = f + (size_t)b * MPTS * 2;
    float mu = 0.0f, sg = 0.0f;
    const float step = 28.0f / 27.0f;
    int n = 0;
    for (int ix = 0; ix < GRIDN; ix++) {
        float gx = -14.0f + step * (float)ix;
        float dx2 = (tx - gx) * (tx - gx);
        for (int iy = 0; iy < GRIDN; iy++, n++) {
            float gy = -14.0f + step * (float)iy;
            float dy = ty - gy;
            float w = osc * __expf(inv * (dx2 + dy * dy));
            mu += w * fb[2 * n];
            sg += w * fb[2 * n + 1];     // fb[...,1] already softplus'd
        }
    }
    out_mu[t] = mu;
    out_sigma[(size_t)b * NTGT * NTGT + (size_t)q * NTGT + q] = sg;
}

// ---------------------------------------------------------------- host
extern "C" void kernel_launch(void* const* d_in, const int* in_sizes, int n_in,
                              void* d_out, int out_size, void* d_ws, size_t ws_size,
                              hipStream_t stream) {
    const float* ctx_coords = (const float*)d_in[0];
    const float* ctx_values = (const float*)d_in[1];
    const float* tgt_coords = (const float*)d_in[2];
    const float* psi_ls = (const float*)d_in[3];
    const float* psi_os = (const float*)d_in[4];
    const float* rho_ls = (const float*)d_in[5];
    const float* rho_os = (const float*)d_in[6];

    ConvW cw;
    for (int li = 0; li < 4; li++) {
        cw.w1[li] = (const float*)d_in[7 + li * 6 + 0];
        cw.b1[li] = (const float*)d_in[7 + li * 6 + 1];
        cw.w2[li] = (const float*)d_in[7 + li * 6 + 2];
        cw.b2[li] = (const float*)d_in[7 + li * 6 + 3];
        cw.lw[li] = (const float*)d_in[7 + li * 6 + 4];
        cw.lb[li] = (const float*)d_in[7 + li * 6 + 5];
    }

    char* ws = (char*)d_ws;
    int*   idx  = (int*)  (ws + 0);            // 784*5*4      = 15,680 B
    float* kern = (float*)(ws + (16  << 10));  // 4*784*5*16*4 ≈ 1.0 MB
    float* W    = (float*)(ws + (1   << 20));  // ≤ 784*160*16*4 ≈ 8.0 MB
    float* vA   = (float*)(ws + (9   << 20));  // 50176*32*4 ≈ 6.4 MB
    float* vB   = (float*)(ws + (16  << 20));  // 50176*32*4 ≈ 6.4 MB

    knn_kernel   <<<(MPTS + 127) / 128, 128, 0, stream>>>(idx);
    encode_kernel<<<(BATCH * MPTS + 255) / 256, 256, 0, stream>>>(
        ctx_coords, ctx_values, psi_ls, psi_os, vA);
    kern_kernel  <<<(4 * MPTS * KNN + 255) / 256, 256, 0, stream>>>(cw, idx, kern);

    const int CIN[4]  = {4, 16, 32, 16};
    const int COUT[4] = {16, 32, 16, 2};
    for (int li = 0; li < 4; li++) {
        int tot = MPTS * KNN * CIN[li] * COUT[li];
        wfold_kernel<<<(tot + 255) / 256, 256, 0, stream>>>(
            kern + (size_t)li * MPTS * KNN * CMCO, cw.lw[li], W, CIN[li], COUT[li]);
        switch (li) {
        case 0: layer_kernel< 4, 16, false><<<MPTS, 128, 0, stream>>>(vA, W, idx, cw.lb[0], vB); break;
        case 1: layer_kernel<16, 32, false><<<MPTS, 128, 0, stream>>>(vB, W, idx, cw.lb[1], vA); break;
        case 2: layer_kernel<32, 16, false><<<MPTS, 128, 0, stream>>>(vA, W, idx, cw.lb[2], vB); break;
        case 3: layer_kernel<16,  2, true ><<<MPTS, 128, 0, stream>>>(vB, W, idx, cw.lb[3], vA); break;
        }
    }
    // after li=3 the result (2 channels: mu, softplus(sigma)) is in vA

    float* mu_out    = (float*)d_out;
    float* sigma_out = mu_out + (size_t)BATCH * NTGT;
    (void)hipMemsetAsync(sigma_out, 0, (size_t)BATCH * NTGT * NTGT * sizeof(float), stream);
    decode_kernel<<<(BATCH * NTGT + 255) / 256, 256, 0, stream>>>(
        tgt_coords, vA, rho_ls, rho_os, mu_out, sigma_out);
}